// Encoder_35253091565790
// MI455X (gfx1250) — compile-verified
//
#include <hip/hip_runtime.h>

typedef float v2f __attribute__((ext_vector_type(2)));
typedef float v8f __attribute__((ext_vector_type(8)));

#define N_TOT_ENT 57344   /* 4096 * (2 + 12) */
#define TILES_ACT 512     /* 8192 active entities / 16 */
#define N_TILES   3584    /* 57344 / 16 */
#define MASK_OFF  14680064 /* 57344 * 256 */

__device__ __forceinline__ void add4(float4& a, const float4 b) {
    a.x += b.x; a.y += b.y; a.z += b.z; a.w += b.w;
}

__global__ __launch_bounds__(128) void entity_encode_kernel(
    const int*   __restrict__ act,  const int*   __restrict__ sid,
    const float* __restrict__ W1h,  const float* __restrict__ bo,
    const float* __restrict__ Wsp,  const float* __restrict__ bsp,
    const float* __restrict__ Wab,  const float* __restrict__ bab,
    const float* __restrict__ Wit,  const float* __restrict__ bit_,
    const float* __restrict__ Wmv,  const float* __restrict__ bmv,
    float* __restrict__ out)
{
    __shared__ int   feats[16][20];
    __shared__ __align__(16) float bsum[256];
    __shared__ float bc[16][68];            // boolean code, K padded to 68
    __shared__ __align__(16) float gsum[16][264]; // 264 stride: conflict-free halves

    const int tid  = threadIdx.x;
    const int tile = blockIdx.x;

    // ---- stage entity features (contiguous 320 ints per tile) ----
    const int* fsrc = (tile < TILES_ACT) ? (act + tile * 320)
                                         : (sid + (tile - TILES_ACT) * 320);
    for (int idx = tid; idx < 320; idx += 128)
        ((int*)feats)[idx] = fsrc[idx];

    // ---- bias sum (b_onehot+b_species+b_ability+b_item+b_moveset) ----
    if (tid < 64) {
        const int c = tid * 4;
        float4 s = *(const float4*)(bo + c);
        add4(s, *(const float4*)(bsp  + c));
        add4(s, *(const float4*)(bab  + c));
        add4(s, *(const float4*)(bit_ + c));
        add4(s, *(const float4*)(bmv  + c));
        *(float4*)&bsum[c] = s;
    }
    __syncthreads();

    // ---- boolean code rows: one thread per entity ----
    if (tid < 16) {
        const int* f = feats[tid];
        float* r = bc[tid];
        #pragma unroll
        for (int j = 0; j < 68; ++j) r[j] = 0.0f;
        const int hp = f[0];
        const int mh = (f[1] < 1) ? 1 : f[1];
        float ratio = (float)hp / (float)mh;
        ratio = fminf(fmaxf(ratio, 0.0f), 1.0f);
        const int tok = (int)(1023.0f * ratio);
        #pragma unroll
        for (int i = 0; i < 10; ++i) r[i] = (float)((tok >> i) & 1);
        r[10] = ratio;
        const int lvl = f[2];
        #pragma unroll
        for (int i = 0; i < 7; ++i) r[11 + i] = (float)((lvl >> i) & 1);
        auto setoh = [&](int base, int v, int n) {
            if (v >= 0 && v < n) r[base + v] = 1.0f;
        };
        setoh(18, f[3], 3);   // gender
        setoh(21, f[4], 8);   // status
        setoh(29, f[5], 2);   // being_called_back
        setoh(31, f[6], 2);   // trapped
        setoh(33, f[7], 2);   // active
        setoh(35, f[8], 2);   // newly_switched
        setoh(37, f[9], 8);   // toxic_turns
        setoh(45, f[10], 4);  // sleep_turns
        setoh(49, f[11], 2);  // fainted
        setoh(51, f[12], 16); // item_effect
    }

    // ---- gather-sum into gsum: 8 threads per entity, float4 lanes ----
    {
        const int e  = tid >> 3;
        const int cb = (tid & 7) * 4;
        const int* f = feats[e];
        const int sp = f[13], ab = f[14], it = f[15];
        const int m0 = f[16], m1 = f[17], m2 = f[18];
        const bool hv0 = (m0 == 0) | (m1 == 0) | (m2 == 0);
        const bool hv1 = (m0 == 1) | (m1 == 1) | (m2 == 1);
        const bool hv2 = (m0 == 2) | (m1 == 2) | (m2 == 2);
        const bool hv3 = (m0 == 3) | (m1 == 3) | (m2 == 3);
        const bool vsp = (sp >= 0) & (sp < 1280);
        const bool vab = (ab >= 0) & (ab < 320);
        const bool vit = (it >= 0) & (it < 1024);
        #pragma unroll
        for (int i = 0; i < 8; ++i) {
            const int c = cb + i * 32;
            float4 acc = *(const float4*)&bsum[c];
            if (vsp) add4(acc, *(const float4*)(Wsp + sp * 256 + c));
            if (vab) add4(acc, *(const float4*)(Wab + ab * 256 + c));
            if (vit) add4(acc, *(const float4*)(Wit + it * 256 + c));
            if (hv0) add4(acc, *(const float4*)(Wmv + 0 * 256 + c));
            if (hv1) add4(acc, *(const float4*)(Wmv + 1 * 256 + c));
            if (hv2) add4(acc, *(const float4*)(Wmv + 2 * 256 + c));
            if (hv3) add4(acc, *(const float4*)(Wmv + 3 * 256 + c));
            *(float4*)&gsum[e][c] = acc;
        }
    }
    __syncthreads();

    // ---- dense part: bc[16x68] @ W_onehot[67x256] via V_WMMA_F32_16X16X4_F32 ----
    const int lane  = tid & 31;
    const int w     = tid >> 5;      // wave id: owns cols [w*64, w*64+64)
    const int lhalf = lane >> 4;     // 0: K+0/K+1, 1: K+2/K+3
    const int l15   = lane & 15;
    v8f cc[4];
    #pragma unroll
    for (int nt = 0; nt < 4; ++nt) cc[nt] = (v8f){0,0,0,0,0,0,0,0};

    const int nbase = w * 64 + l15;
    for (int ks = 0; ks < 17; ++ks) {
        const int ka = ks * 4 + lhalf * 2;            // <= 66
        v2f a;
        a.x = bc[l15][ka];
        a.y = bc[l15][ka + 1];                         // [67] is the zero pad
        const int rB = (ka + 1 > 66) ? 66 : (ka + 1);  // clamp OOB row; A is 0 there
        const float* pA = W1h + ka * 256 + nbase;
        const float* pB = W1h + rB * 256 + nbase;
        #pragma unroll
        for (int nt = 0; nt < 4; ++nt) {
            v2f b;
            b.x = pA[nt * 16];
            b.y = pB[nt * 16];
            cc[nt] = __builtin_amdgcn_wmma_f32_16x16x4_f32(
                false, a, false, b, (short)0, cc[nt], false, false);
        }
    }

    // ---- merge WMMA accumulators into gsum (exclusive ownership) ----
    #pragma unroll
    for (int nt = 0; nt < 4; ++nt) {
        const int ncol = w * 64 + nt * 16 + l15;
        const int mb   = lhalf * 8;
        #pragma unroll
        for (int r = 0; r < 8; ++r)
            gsum[mb + r][ncol] += cc[nt][r];
    }
    __syncthreads();

    // ---- coalesced float4 writeout ----
    {
        const int e = tid >> 3;
        const long long ge = (long long)tile * 16 + e;
        float* op = out + ge * 256;
        #pragma unroll
        for (int i = 0; i < 8; ++i) {
            const int c = (tid & 7) * 4 + i * 32;
            *(float4*)&op[c] = *(const float4*)&gsum[e][c];
        }
    }

    // ---- valid_team_mask: always true per reference semantics ----
    if (tile >= TILES_ACT && tid < 16)
        out[MASK_OFF + (tile - TILES_ACT) * 16 + tid] = 1.0f;
}

extern "C" void kernel_launch(void* const* d_in, const int* in_sizes, int n_in,
                              void* d_out, int out_size, void* d_ws, size_t ws_size,
                              hipStream_t stream) {
    (void)in_sizes; (void)n_in; (void)out_size; (void)d_ws; (void)ws_size;
    const int*   act  = (const int*)d_in[0];
    const int*   sid  = (const int*)d_in[1];
    const float* W1h  = (const float*)d_in[2];
    const float* bo   = (const float*)d_in[3];
    const float* Wsp  = (const float*)d_in[4];
    const float* bsp  = (const float*)d_in[5];
    const float* Wab  = (const float*)d_in[6];
    const float* bab  = (const float*)d_in[7];
    const float* Wit  = (const float*)d_in[8];
    const float* bit_ = (const float*)d_in[9];
    const float* Wmv  = (const float*)d_in[10];
    const float* bmv  = (const float*)d_in[11];
    float* out = (float*)d_out;

    entity_encode_kernel<<<dim3(N_TILES), dim3(128), 0, stream>>>(
        act, sid, W1h, bo, Wsp, bsp, Wab, bab, Wit, bit_, Wmv, bmv, out);
}